// LSTMLayer_31396210934320
// MI455X (gfx1250) — compile-verified
//
#include <hip/hip_runtime.h>
#include <math.h>

typedef __attribute__((ext_vector_type(2))) float v2f;
typedef __attribute__((ext_vector_type(8))) float v8f;

#define T_LEN   2048
#define IN_CH   256
#define CH      256      // channels per gate
#define KDIM    512      // IN_CH * 2 taps
#define KBLK    64       // K per LDS stage (32 input channels)
#define ICBLK   32
#define NCHUNK  128      // output columns (t) per workgroup
#define XSTR    132      // padded LDS stride for x slice (129 used)
#define ASTR    68       // padded LDS stride for W rows (64 used)

__global__ __launch_bounds__(256)
void lstm_fused_wmma(const float* __restrict__ x,
                     const float* __restrict__ W,
                     const float* __restrict__ bias,
                     float* __restrict__ out)
{
    __shared__ float A_lds[3 * 16 * ASTR];   // 3 gates x 16 rows x 64 k (padded)
    __shared__ float x_lds[ICBLK * XSTR];    // 32 ic x 129 t (padded), t0-1 .. t0+127

    const int tid   = threadIdx.x;
    const int lane  = tid & 31;
    const int wave  = tid >> 5;
    const int half  = lane >> 4;     // 0: lanes 0-15, 1: lanes 16-31
    const int l15   = lane & 15;
    const int cbase = blockIdx.x * 16;           // gate-channel tile base (0..255)
    const int chunk = blockIdx.y;                // 64 chunks total
    const int b     = chunk >> 4;                // 16 chunks per batch (T/128)
    const int t0    = (chunk & 15) * NCHUNK;
    const int colw  = wave * 16;                 // wave's column base within chunk

    v8f accz = {}; v8f accf = {}; v8f acco = {};

    for (int kb = 0; kb < KDIM; kb += KBLK) {
        __syncthreads();
        const int ic0 = kb >> 1;

        // Stage W rows for gates z,f,o: A[k] with k = ic*2 + tap (contiguous in W)
        for (int idx = tid; idx < 3 * 16 * 16; idx += 256) {
            int g   = idx >> 8;          // gate 0..2
            int rem = idx & 255;
            int m   = rem >> 4;          // row 0..15
            int kq  = (rem & 15) << 2;   // k quad 0,4,...,60
            float4 w4 = *(const float4*)(W + (size_t)((g * CH + cbase + m) * KDIM + kb + kq));
            *(float4*)(A_lds + (g * 16 + m) * ASTR + kq) = w4;
        }

        // Stage x slice: x_lds[ic][j] = x[b, ic0+ic, t0-1+j], j in [0,128], pad t=-1 -> 0
        for (int idx = tid; idx < ICBLK * XSTR; idx += 256) {
            int ic = idx / XSTR;
            int jj = idx - ic * XSTR;
            if (jj <= NCHUNK) {
                int t = t0 - 1 + jj;
                float v = 0.0f;
                if (t >= 0)
                    v = x[(size_t)((b * IN_CH + ic0 + ic) * T_LEN + t)];
                x_lds[idx] = v;
            }
        }
        __syncthreads();

        #pragma unroll
        for (int ks = 0; ks < KBLK / 4; ++ks) {
            // A fragment k indices for this lane half: kbase + {0,1} or {2,3}
            const int k  = ks * 4 + half * 2;
            // B fragment: rows k,k+1 of im2col = taps 0,1 of ic = two adjacent t values
            const int ic  = ks * 2 + half;
            const int col = colw + l15;
            v2f bf;
            bf.x = x_lds[ic * XSTR + col];       // x[.., t-1] (tap 0)
            bf.y = x_lds[ic * XSTR + col + 1];   // x[.., t]   (tap 1)

            v2f a;
            const float* az = A_lds + (0 * 16 + l15) * ASTR + k;
            a.x = az[0]; a.y = az[1];
            accz = __builtin_amdgcn_wmma_f32_16x16x4_f32(false, a, false, bf,
                                                         (short)0, accz, false, false);
            const float* af = A_lds + (1 * 16 + l15) * ASTR + k;
            a.x = af[0]; a.y = af[1];
            accf = __builtin_amdgcn_wmma_f32_16x16x4_f32(false, a, false, bf,
                                                         (short)0, accf, false, false);
            const float* ao = A_lds + (2 * 16 + l15) * ASTR + k;
            a.x = ao[0]; a.y = ao[1];
            acco = __builtin_amdgcn_wmma_f32_16x16x4_f32(false, a, false, bf,
                                                         (short)0, acco, false, false);
        }
    }

    // Epilogue: bias, activations, closed-form "scan", h = o * c
    const int t = t0 + colw + l15;
    #pragma unroll
    for (int r = 0; r < 8; ++r) {
        int m = r + half * 8;                       // C/D layout: VGPR r holds M=r / M=r+8
        float bz = bias[0 * CH + cbase + m];
        float bfv = bias[1 * CH + cbase + m];
        float bo = bias[2 * CH + cbase + m];

        float z  = tanhf(accz[r] + bz);
        float f  = 1.0f / (1.0f + expf(-(accf[r] + bfv)));
        float og = 1.0f / (1.0f + expf(-(acco[r] + bo)));

        // c_final = z * S(f), S = ((n-1) - n*f + f^n) / (1-f)^2, n = T_LEN
        double df = (double)f;
        double fn = df;
        #pragma unroll
        for (int q = 0; q < 11; ++q) fn *= fn;      // f^2048
        double omf = 1.0 - df;
        double S;
        if (omf > 1e-12)
            S = ((double)(T_LEN - 1) - (double)T_LEN * df + fn) / (omf * omf);
        else
            S = 0.5 * (double)T_LEN * (double)(T_LEN - 1);

        float c = z * (float)S;
        out[(size_t)((b * CH + cbase + m) * T_LEN + t)] = og * c;
    }
}

extern "C" void kernel_launch(void* const* d_in, const int* in_sizes, int n_in,
                              void* d_out, int out_size, void* d_ws, size_t ws_size,
                              hipStream_t stream) {
    const float* x    = (const float*)d_in[0];   // [4, 256, 2048]
    const float* W    = (const float*)d_in[1];   // [1024, 256, 2]
    const float* bias = (const float*)d_in[2];   // [1024]
    float* out = (float*)d_out;                  // [4, 256, 2048]

    dim3 grid(CH / 16, (4 * T_LEN) / NCHUNK);    // 16 x 64
    lstm_fused_wmma<<<grid, 256, 0, stream>>>(x, W, bias, out);
}